// SpatialGATLayer_43473658970662
// MI455X (gfx1250) — compile-verified
//
#include <hip/hip_runtime.h>
#include <hip/hip_bf16.h>
#include <math.h>

typedef float v2f __attribute__((ext_vector_type(2)));
typedef float v8f __attribute__((ext_vector_type(8)));

#define NN 20000
#define EE 640000
#define IN_DIM 256
#define HEADS 4
#define CH 32
#define HC 128
#define NEG_SLOPE 0.2f
#define LN_EPS 1e-5f

// ---------------------------------------------------------------------------
// Kernel 1: xl = x@Wl + bl ; xr = x@Wr + br   (fp32 WMMA 16x16x4)
// block = 512 threads = 16 waves. wave>>3 selects {Wl,Wr}; wave&7 selects the
// 16-wide N tile (HC=128 = 8 tiles). blockIdx.x selects 16 M rows (N=20000 is
// divisible by 16).
// ---------------------------------------------------------------------------
__global__ __launch_bounds__(512) void k_lin(
    const float* __restrict__ x,
    const float* __restrict__ Wl, const float* __restrict__ bl,
    const float* __restrict__ Wr, const float* __restrict__ br,
    float* __restrict__ xl, float* __restrict__ xr)
{
    const int wave = threadIdx.x >> 5;
    const int lane = threadIdx.x & 31;
    const int half = lane >> 4;
    const int l    = lane & 15;
    const int m0   = blockIdx.x * 16;
    const int sel  = wave >> 3;
    const int n0   = (wave & 7) * 16;

    const float* __restrict__ W   = sel ? Wr : Wl;
    const float* __restrict__ bia = sel ? br : bl;
    float* __restrict__ out       = sel ? xr : xl;

    const int row = m0 + l;
    const float* __restrict__ xrow = x + (size_t)row * IN_DIM + half * 2;
    const float* __restrict__ wcol = W + (size_t)(half * 2) * HC + n0 + l;

    v8f acc = {};
    #pragma unroll 4
    for (int k = 0; k < IN_DIM; k += 4) {
        v2f a, b;
        a.x = xrow[k];
        a.y = xrow[k + 1];
        b.x = wcol[(size_t)k * HC];
        b.y = wcol[(size_t)k * HC + HC];
        acc = __builtin_amdgcn_wmma_f32_16x16x4_f32(
            false, a, false, b, (short)0, acc, false, false);
    }

    const float bb = bia[n0 + l];
    #pragma unroll
    for (int r = 0; r < 8; ++r) {
        out[(size_t)(m0 + r + half * 8) * HC + n0 + l] = acc[r] + bb;
    }
}

// ---------------------------------------------------------------------------
// Kernel 2: init m = -inf, denom = 0, agg = 0
// ---------------------------------------------------------------------------
__global__ void k_init(float* __restrict__ m, float* __restrict__ denom,
                       float* __restrict__ agg, int nm, int nagg)
{
    int i = blockIdx.x * blockDim.x + threadIdx.x;
    if (i < nagg) agg[i] = 0.0f;
    if (i < nm) { m[i] = -INFINITY; denom[i] = 0.0f; }
}

// float atomic max via sign-aware integer atomics (works for mixed signs,
// init = -inf)
__device__ inline void atomicMaxF(float* addr, float v)
{
    if (v >= 0.0f) atomicMax((int*)addr, __float_as_int(v));
    else           atomicMin((unsigned int*)addr, __float_as_uint(v));
}

// ---------------------------------------------------------------------------
// Kernel 3: per-edge logits + segment max.  One wave per edge; lane j holds
// flat channels 4j..4j+3 (head = j>>3). Reduce within 8-lane head groups.
// ---------------------------------------------------------------------------
__global__ __launch_bounds__(256) void k_logits(
    const int* __restrict__ src, const int* __restrict__ dst,
    const float* __restrict__ xl, const float* __restrict__ xr,
    const float* __restrict__ att,
    float* __restrict__ logits, float* __restrict__ m)
{
    const int e = (blockIdx.x * blockDim.x + threadIdx.x) >> 5;
    const int lane = threadIdx.x & 31;
    if (e >= EE) return;

    const int s = src[e];
    const int d = dst[e];

    const float4 a  = ((const float4*)att)[lane];
    const float4 vs = ((const float4*)(xl + (size_t)s * HC))[lane];
    const float4 vd = ((const float4*)(xr + (size_t)d * HC))[lane];

    float p = 0.0f;
    {
        float t;
        t = vs.x + vd.x; t = (t > 0.0f) ? t : t * NEG_SLOPE; p += t * a.x;
        t = vs.y + vd.y; t = (t > 0.0f) ? t : t * NEG_SLOPE; p += t * a.y;
        t = vs.z + vd.z; t = (t > 0.0f) ? t : t * NEG_SLOPE; p += t * a.z;
        t = vs.w + vd.w; t = (t > 0.0f) ? t : t * NEG_SLOPE; p += t * a.w;
    }
    // reduce within the 8-lane group owning one head
    p += __shfl_xor(p, 1, 32);
    p += __shfl_xor(p, 2, 32);
    p += __shfl_xor(p, 4, 32);

    if ((lane & 7) == 0) {
        const int h = lane >> 3;
        logits[(size_t)e * HEADS + h] = p;
        atomicMaxF(&m[(size_t)d * HEADS + h], p);
    }
}

// ---------------------------------------------------------------------------
// Kernel 4: m = where(finite, m, 0)
// ---------------------------------------------------------------------------
__global__ void k_fixm(float* __restrict__ m, int n)
{
    int i = blockIdx.x * blockDim.x + threadIdx.x;
    if (i < n) {
        float v = m[i];
        if (!isfinite(v)) m[i] = 0.0f;
    }
}

// ---------------------------------------------------------------------------
// Kernel 5: expv = exp(logit - m[dst]) (in place over logits) + denom atomics
// ---------------------------------------------------------------------------
__global__ void k_exp(const int* __restrict__ dst,
                      float* __restrict__ logits,   // in: logits, out: expv
                      const float* __restrict__ m,
                      float* __restrict__ denom)
{
    int idx = blockIdx.x * blockDim.x + threadIdx.x;
    if (idx >= EE * HEADS) return;
    const int e = idx >> 2;
    const int h = idx & 3;
    const int d = dst[e];
    const float v = expf(logits[idx] - m[(size_t)d * HEADS + h]);
    logits[idx] = v;
    atomicAdd(&denom[(size_t)d * HEADS + h], v);
}

// ---------------------------------------------------------------------------
// Kernel 6: agg[dst] += xl[src] * alpha.  Wave per edge, 4 channels/lane.
// ---------------------------------------------------------------------------
__global__ __launch_bounds__(256) void k_scatter(
    const int* __restrict__ src, const int* __restrict__ dst,
    const float* __restrict__ xl,
    const float* __restrict__ expv, const float* __restrict__ denom,
    float* __restrict__ agg)
{
    const int e = (blockIdx.x * blockDim.x + threadIdx.x) >> 5;
    const int lane = threadIdx.x & 31;
    if (e >= EE) return;

    const int s = src[e];
    const int d = dst[e];
    const int h = lane >> 3;

    const float alpha = expv[(size_t)e * HEADS + h] /
                        (denom[(size_t)d * HEADS + h] + 1e-16f);

    const float4 v = ((const float4*)(xl + (size_t)s * HC))[lane];
    float* base = agg + (size_t)d * HC + lane * 4;
    atomicAdd(base + 0, v.x * alpha);
    atomicAdd(base + 1, v.y * alpha);
    atomicAdd(base + 2, v.z * alpha);
    atomicAdd(base + 3, v.w * alpha);
}

// ---------------------------------------------------------------------------
// Kernel 7: h = x + (agg + bias) @ proj_W + proj_b ; LayerNorm(h)
// block = 512 threads = 16 waves. Wave w computes the 16-col tile n0=16w, so
// one block produces 16 complete 256-wide rows -> LayerNorm fused via LDS.
// ---------------------------------------------------------------------------
__global__ __launch_bounds__(512) void k_proj_ln(
    const float* __restrict__ x,
    const float* __restrict__ agg, const float* __restrict__ bias,
    const float* __restrict__ projW, const float* __restrict__ projb,
    const float* __restrict__ ln_g, const float* __restrict__ ln_b,
    float* __restrict__ outp)
{
    __shared__ float hbuf[16 * IN_DIM];

    const int wave = threadIdx.x >> 5;
    const int lane = threadIdx.x & 31;
    const int half = lane >> 4;
    const int l    = lane & 15;
    const int m0   = blockIdx.x * 16;
    const int n0   = wave * 16;

    const int row = m0 + l;
    const float* __restrict__ arow = agg + (size_t)row * HC + half * 2;
    const float* __restrict__ brow = bias + half * 2;
    const float* __restrict__ wcol = projW + (size_t)(half * 2) * IN_DIM + n0 + l;

    v8f acc = {};
    #pragma unroll 4
    for (int k = 0; k < HC; k += 4) {
        v2f a, b;
        a.x = arow[k]     + brow[k];
        a.y = arow[k + 1] + brow[k + 1];
        b.x = wcol[(size_t)k * IN_DIM];
        b.y = wcol[(size_t)k * IN_DIM + IN_DIM];
        acc = __builtin_amdgcn_wmma_f32_16x16x4_f32(
            false, a, false, b, (short)0, acc, false, false);
    }

    const float pb = projb[n0 + l];
    #pragma unroll
    for (int r = 0; r < 8; ++r) {
        const int rr = r + half * 8;
        hbuf[rr * IN_DIM + n0 + l] =
            acc[r] + pb + x[(size_t)(m0 + rr) * IN_DIM + n0 + l];
    }
    __syncthreads();

    // LayerNorm: wave `wave` normalizes row `wave` (16 waves, 16 rows)
    const int rowi = wave;
    float s = 0.0f, s2 = 0.0f;
    #pragma unroll
    for (int j = lane; j < IN_DIM; j += 32) {
        const float v = hbuf[rowi * IN_DIM + j];
        s += v; s2 += v * v;
    }
    #pragma unroll
    for (int off = 16; off > 0; off >>= 1) {
        s  += __shfl_xor(s,  off, 32);
        s2 += __shfl_xor(s2, off, 32);
    }
    const float mu  = s * (1.0f / IN_DIM);
    const float var = s2 * (1.0f / IN_DIM) - mu * mu;
    const float inv = rsqrtf(var + LN_EPS);

    #pragma unroll
    for (int j = lane; j < IN_DIM; j += 32) {
        const float v = (hbuf[rowi * IN_DIM + j] - mu) * inv;
        outp[(size_t)(m0 + rowi) * IN_DIM + j] = v * ln_g[j] + ln_b[j];
    }
}

// ---------------------------------------------------------------------------
extern "C" void kernel_launch(void* const* d_in, const int* in_sizes, int n_in,
                              void* d_out, int out_size, void* d_ws, size_t ws_size,
                              hipStream_t stream)
{
    const float* x      = (const float*)d_in[0];
    const int*   eidx   = (const int*)  d_in[1];   // [2, E]
    const float* Wl     = (const float*)d_in[2];
    const float* bl     = (const float*)d_in[3];
    const float* Wr     = (const float*)d_in[4];
    const float* br     = (const float*)d_in[5];
    const float* att    = (const float*)d_in[6];
    const float* bias   = (const float*)d_in[7];
    const float* projW  = (const float*)d_in[8];
    const float* projb  = (const float*)d_in[9];
    const float* ln_g   = (const float*)d_in[10];
    const float* ln_b   = (const float*)d_in[11];
    float* outp = (float*)d_out;

    const int* src = eidx;
    const int* dst = eidx + EE;

    // workspace layout (floats)
    float* ws     = (float*)d_ws;
    float* xl     = ws;                          // N*HC
    float* xr     = xl + (size_t)NN * HC;        // N*HC
    float* logits = xr + (size_t)NN * HC;        // E*HEADS (reused as expv)
    float* mbuf   = logits + (size_t)EE * HEADS; // N*HEADS
    float* denom  = mbuf + (size_t)NN * HEADS;   // N*HEADS
    float* agg    = denom + (size_t)NN * HEADS;  // N*HC

    // 1) xl, xr GEMMs (WMMA fp32)
    k_lin<<<NN / 16, 512, 0, stream>>>(x, Wl, bl, Wr, br, xl, xr);

    // 2) init reductions buffers
    {
        const int nagg = NN * HC;
        const int nm   = NN * HEADS;
        k_init<<<(nagg + 255) / 256, 256, 0, stream>>>(mbuf, denom, agg, nm, nagg);
    }

    // 3) per-edge logits + segment max (wave per edge)
    k_logits<<<EE / 8, 256, 0, stream>>>(src, dst, xl, xr, att, logits, mbuf);

    // 4) sanitize max for isolated nodes
    k_fixm<<<(NN * HEADS + 255) / 256, 256, 0, stream>>>(mbuf, NN * HEADS);

    // 5) exp + denom
    k_exp<<<(EE * HEADS + 255) / 256, 256, 0, stream>>>(dst, logits, mbuf, denom);

    // 6) weighted scatter-sum (wave per edge)
    k_scatter<<<EE / 8, 256, 0, stream>>>(src, dst, xl, logits, denom, agg);

    // 7) projection + residual + LayerNorm (WMMA fp32, fused LN)
    k_proj_ln<<<NN / 16, 512, 0, stream>>>(x, agg, bias, projW, projb,
                                           ln_g, ln_b, outp);
}